// BrightnessImportanceSampler_45938970198209
// MI455X (gfx1250) — compile-verified
//
#include <hip/hip_runtime.h>
#include <hip/hip_bf16.h>

// BrightnessImportanceSampler for MI455X (gfx1250, wave32).
//
// Strategy (bandwidth-bound: ~268 MB of output @ 23.3 TB/s => ~11.5us floor):
//  - Invert the reference's scatter (t = num_ray-1-j, injective) into a gather:
//    one coalesced streaming write pass, every output element written once.
//  - Per 16-ray tile, compute the dense base dot table spots(512x3) . N^T(3x16)
//    with V_WMMA_F32_16X16X4_F32 (K=3 zero-padded to 4) into LDS.
//  - Bit-exact JAX threefry2x32 RNG (key(42)) for the jitter uniforms, only on
//    candidate elements (j in [0,num_ele)).
//  - Prefix-mask row counts reduced via LDS atomics (ds_add_u32).

typedef __attribute__((ext_vector_type(2))) float v2f;
typedef __attribute__((ext_vector_type(8))) float v8f;

#define HAS_WMMA_F32_16X16X4 __has_builtin(__builtin_amdgcn_wmma_f32_16x16x4_f32)

constexpr int Bn = 32768;
constexpr int Sn = 512;
constexpr int RAYS_PER_BLOCK = 16;
constexpr int THREADS = 256;   // 8 wave32s

// ---------------- JAX threefry2x32 (seed 42) ----------------
__device__ __forceinline__ unsigned rotl32(unsigned x, int r) {
    return (x << r) | (x >> (32 - r));
}

// Returns out0 (want_hi=false) or out1 (want_hi=true) of threefry2x32 with
// key (0, 42) applied to counter pair (c0, c1).
__device__ __forceinline__ unsigned tf2x32(unsigned x0, unsigned x1, bool want_hi) {
    const unsigned k0 = 0u, k1 = 42u;
    const unsigned k2 = 0x1BD11BDAu ^ k0 ^ k1;
    const unsigned ks[3] = {k0, k1, k2};
    const int rotA[4] = {13, 15, 26, 6};
    const int rotB[4] = {17, 29, 16, 24};
    x0 += ks[0];
    x1 += ks[1];
#pragma unroll
    for (int g = 0; g < 5; ++g) {
        const int* rot = (g & 1) ? rotB : rotA;
#pragma unroll
        for (int i = 0; i < 4; ++i) {
            x0 += x1;
            x1 = rotl32(x1, rot[i]);
            x1 ^= x0;
        }
        x0 += ks[(g + 1) % 3];
        x1 += ks[(g + 2) % 3] + (unsigned)(g + 1);
    }
    return want_hi ? x1 : x0;
}

// JAX random.uniform element e of a flat array of n elements (H = n/2):
// element e < H pairs with (e, e+H) -> out0; else (e-H, e) -> out1.
__device__ __forceinline__ float jax_u01(unsigned e, unsigned H) {
    const unsigned bits = (e < H) ? tf2x32(e, e + H, false)
                                  : tf2x32(e - H, e, true);
    const unsigned f = 0x3f800000u | (bits >> 9);
    return __uint_as_float(f) - 1.0f;
}

// ---------------- Kernel ----------------
__global__ __launch_bounds__(THREADS)
void bis_kernel(const float* __restrict__ Nrm,
                const float* __restrict__ spots,
                const int* __restrict__ ray_mask,
                const int* __restrict__ bright_mask,
                const float* __restrict__ std_p,
                float* __restrict__ out) {
    __shared__ float s_spots[Sn * 3];                 // 6 KB
    __shared__ float s_base[Sn][RAYS_PER_BLOCK];      // 32 KB: spots . N
    __shared__ float s_N[RAYS_PER_BLOCK][3];
    __shared__ int   s_nray[RAYS_PER_BLOCK];
    __shared__ int   s_nele[RAYS_PER_BLOCK];

    const int tid = threadIdx.x;
    const int b0  = blockIdx.x * RAYS_PER_BLOCK;
    const float stdv = std_p[0];

    // ---- Stage 0: stage spots + N tile into LDS; init counters ----
    for (int i = tid; i < Sn * 3; i += THREADS) s_spots[i] = spots[i];
    if (tid < RAYS_PER_BLOCK) { s_nray[tid] = 0; s_nele[tid] = 0; }
    if (tid < RAYS_PER_BLOCK * 3) s_N[tid / 3][tid % 3] = Nrm[b0 * 3 + tid];
    __syncthreads();

    // ---- Stage 0b: row counts of the prefix masks (16 threads per row) ----
    {
        const int r = tid >> 4;
        const int c = tid & 15;
        const int base = (b0 + r) * Sn + c * 32;
        int sr = 0, se = 0;
#pragma unroll 4
        for (int k = 0; k < 32; ++k) {
            sr += ray_mask[base + k];
            se += bright_mask[base + k];
        }
        atomicAdd(&s_nray[r], sr);   // ds_add_u32
        atomicAdd(&s_nele[r], se);
    }

    // ---- Stage 1: base[s][r] = spots[s] . N[b0+r] via WMMA f32 16x16x4 ----
    // Wave w covers s in [w*64, w*64+64): 4 WMMAs/wave, 32 WMMAs/block.
    const int wave = tid >> 5;
    const int lane = tid & 31;
#if HAS_WMMA_F32_16X16X4
    {
        const int hi16 = lane >> 4;   // 0: lanes 0-15, 1: lanes 16-31
        const int l16  = lane & 15;
        // B operand (4x16, KxN): VGPR0 = {K=0 | K=1}, VGPR1 = {K=2 | K=3(pad)}
        v2f bmat;
        bmat.x = hi16 ? s_N[l16][1] : s_N[l16][0];
        bmat.y = hi16 ? 0.0f        : s_N[l16][2];
#pragma unroll
        for (int st = 0; st < 4; ++st) {
            const int s0 = (wave * 4 + st) * 16;
            // A operand (16x4, MxK): VGPR0 = {K=0 | K=2}, VGPR1 = {K=1 | K=3(pad)}
            v2f amat;
            amat.x = hi16 ? s_spots[(s0 + l16) * 3 + 2] : s_spots[(s0 + l16) * 3 + 0];
            amat.y = hi16 ? 0.0f                        : s_spots[(s0 + l16) * 3 + 1];
            v8f c = {0.f, 0.f, 0.f, 0.f, 0.f, 0.f, 0.f, 0.f};
            v8f d = __builtin_amdgcn_wmma_f32_16x16x4_f32(
                false, amat, false, bmat, (short)0, c, false, false);
            // C/D 16x16: VGPR i holds M=i (lanes 0-15) / M=8+i (lanes 16-31), N=lane%16
            const int mbase = s0 + (hi16 ? 8 : 0);
#pragma unroll
            for (int i = 0; i < 8; ++i) s_base[mbase + i][l16] = d[i];
        }
    }
#else
    // Scalar fallback: 3-FMA dot per (s, r)
    for (int p = tid; p < Sn * RAYS_PER_BLOCK; p += THREADS) {
        const int s = p >> 4, r = p & 15;
        s_base[s][r] = s_spots[s * 3 + 0] * s_N[r][0] +
                       s_spots[s * 3 + 1] * s_N[r][1] +
                       s_spots[s * 3 + 2] * s_N[r][2];
    }
    (void)wave; (void)lane;
#endif
    __syncthreads();

    // ---- Stage 2: gather pass — one coalesced write per output element ----
    const unsigned H = (unsigned)Bn * (unsigned)Sn * 3u / 2u;  // n/2 = 25165824
    float* __restrict__ Ls = out;                               // (B,S,3)
    float* __restrict__ bm = out + (size_t)Bn * Sn * 3;         // (B,S)

    for (int p = tid; p < RAYS_PER_BLOCK * Sn; p += THREADS) {
        const int r = p >> 9;          // / Sn
        const int t = p & (Sn - 1);
        const int b = b0 + r;
        const int j = s_nray[r] - 1 - t;   // inverse of t = num_ray-1-j
        float ox = 0.f, oy = 0.f, oz = 0.f, om = 0.f;
        if (j >= 0 && j < s_nele[r]) {
            const unsigned e = ((unsigned)b * (unsigned)Sn + (unsigned)j) * 3u;
            const float u0 = jax_u01(e + 0u, H);
            const float u1 = jax_u01(e + 1u, H);
            const float u2 = jax_u01(e + 2u, H);
            const float jx = s_spots[j * 3 + 0] + stdv * u0;
            const float jy = s_spots[j * 3 + 1] + stdv * u1;
            const float jz = s_spots[j * 3 + 2] + stdv * u2;
            const float ldotn = s_base[j][r] +
                stdv * (u0 * s_N[r][0] + u1 * s_N[r][1] + u2 * s_N[r][2]);
            if (ldotn > 0.0f) { ox = jx; oy = jy; oz = jz; om = 1.f; }
        }
        const size_t o = (size_t)b * Sn + t;
        Ls[o * 3 + 0] = ox;
        Ls[o * 3 + 1] = oy;
        Ls[o * 3 + 2] = oz;
        bm[o] = om;
    }
}

extern "C" void kernel_launch(void* const* d_in, const int* in_sizes, int n_in,
                              void* d_out, int out_size, void* d_ws, size_t ws_size,
                              hipStream_t stream) {
    // Inputs (setup_inputs order): V(unused), N, spots, ray_mask, bright_mask, std
    const float* Nrm         = (const float*)d_in[1];
    const float* spots       = (const float*)d_in[2];
    const int*   ray_mask    = (const int*)d_in[3];
    const int*   bright_mask = (const int*)d_in[4];
    const float* std_p       = (const float*)d_in[5];
    float* out = (float*)d_out;   // [B*S*3 Ls | B*S mask]

    dim3 grid(Bn / RAYS_PER_BLOCK);   // 2048 blocks x 16 rays
    bis_kernel<<<grid, THREADS, 0, stream>>>(Nrm, spots, ray_mask, bright_mask,
                                             std_p, out);
    (void)in_sizes; (void)n_in; (void)out_size; (void)d_ws; (void)ws_size;
}